// GraphSAGEFraudDetector_7584912245133
// MI455X (gfx1250) — compile-verified
//
#include <hip/hip_runtime.h>

// CDNA5 / gfx1250, wave32. Uses V_WMMA_F32_16X16X4_F32 (full fp32 tensor op).
typedef __attribute__((ext_vector_type(2))) float v2f;
typedef __attribute__((ext_vector_type(8))) float v8f;

#define NNODES 100000
#define HID 64

// ---------------------------------------------------------------- zero fill
__global__ void zero_f32(float* __restrict__ p, size_t n) {
  size_t i = (size_t)blockIdx.x * blockDim.x + threadIdx.x;
  size_t stride = (size_t)gridDim.x * blockDim.x;
  for (; i < n; i += stride) p[i] = 0.0f;
}

// ------------------------------------------------- edge scatter (sum + deg)
// One thread per edge. Float atomics execute at L2 (sums buffer is L2
// resident on MI455X: <= 25.6MB vs 192MB L2), no-return form -> STOREcnt.
template <int K, bool COUNT>
__global__ void __launch_bounds__(256)
scatter_sum(const float* __restrict__ feat,
            const int* __restrict__ src,
            const int* __restrict__ dst,
            float* __restrict__ sums,
            float* __restrict__ cnt,
            int nE) {
  int e = blockIdx.x * blockDim.x + threadIdx.x;
  if (e >= nE) return;
  int s = src[e];
  int d = dst[e];
  if (COUNT) atomicAdd(&cnt[d], 1.0f);
  const float4* fs = (const float4*)(feat + (size_t)s * K);
  float* os = sums + (size_t)d * K;
#pragma unroll
  for (int k4 = 0; k4 < K / 4; ++k4) {
    float4 v = fs[k4];
    atomicAdd(&os[4 * k4 + 0], v.x);
    atomicAdd(&os[4 * k4 + 1], v.y);
    atomicAdd(&os[4 * k4 + 2], v.z);
    atomicAdd(&os[4 * k4 + 3], v.w);
  }
}

// ---------------------------------------------- fused SAGE layer via WMMA
// out[16 rows x 64 cols] per block; 4 waves, one 16x16 column tile per wave.
// D = (sums/deg) @ Wl + x @ Wr + bl, then ReLU.  fp32 WMMA 16x16x4.
//
// A 16x4 f32 layout: lane m=lane&15 holds row m; VGPR0/1 = K={0,1} for lanes
// 0-15, K={2,3} for lanes 16-31.  B 4x16: VGPR0/1 hold rows K={0,1} (lanes
// 0-15, N=lane) and K={2,3} (lanes 16-31, N=lane-16).  C/D: VGPRi = row i
// (lanes 0-15) / row i+8 (lanes 16-31), N=lane&15.
template <int K>
__global__ void __launch_bounds__(128)
sage_layer_wmma(const float* __restrict__ xfeat,  // [N,K] self features
                const float* __restrict__ sums,   // [N,K] neighbor sums
                const float* __restrict__ cnt,    // [N] degrees (float)
                const float* __restrict__ Wl,     // [K,64]
                const float* __restrict__ Wr,     // [K,64]
                const float* __restrict__ bl,     // [64]
                float* __restrict__ out) {        // [N,64]
  const int lane = threadIdx.x & 31;
  const int wave = threadIdx.x >> 5;   // 0..3 -> column tile of 16
  const int m    = lane & 15;
  const int kh   = lane >> 4;          // half-wave selects K pair / row half
  const int row  = blockIdx.x * 16 + m;      // A-matrix row for this lane
  const int col  = wave * 16 + m;            // C/D column for this lane

  const float invCnt = 1.0f / fmaxf(cnt[row], 1.0f);

  const float* sRow = sums  + (size_t)row * K;
  const float* xRow = xfeat + (size_t)row * K;

  v8f acc = {};

  // mean(x_neighbors) @ Wl  (scale-by-1/deg fused into the A operand)
#pragma unroll
  for (int k0 = 0; k0 < K; k0 += 4) {
    const int ka = k0 + kh * 2;
    v2f a, b;
    a.x = sRow[ka]     * invCnt;
    a.y = sRow[ka + 1] * invCnt;
    b.x = Wl[(size_t)ka * HID + col];
    b.y = Wl[(size_t)(ka + 1) * HID + col];
    acc = __builtin_amdgcn_wmma_f32_16x16x4_f32(false, a, false, b,
                                                (short)0, acc, false, false);
  }
  // x_self @ Wr accumulated into same tile
#pragma unroll
  for (int k0 = 0; k0 < K; k0 += 4) {
    const int ka = k0 + kh * 2;
    v2f a, b;
    a.x = xRow[ka];
    a.y = xRow[ka + 1];
    b.x = Wr[(size_t)ka * HID + col];
    b.y = Wr[(size_t)(ka + 1) * HID + col];
    acc = __builtin_amdgcn_wmma_f32_16x16x4_f32(false, a, false, b,
                                                (short)0, acc, false, false);
  }

  // bias + ReLU epilogue; C/D row = i (+8 for upper half-wave)
  const float bias  = bl[col];
  const int   rBase = blockIdx.x * 16 + kh * 8;
#pragma unroll
  for (int i = 0; i < 8; ++i) {
    float v = acc[i] + bias;
    out[(size_t)(rBase + i) * HID + col] = fmaxf(v, 0.0f);
  }
}

// --------------------------------------------------- link-prediction head
// logits[e] = dot(h2[s], Wh[0:64]) + dot(h2[d], Wh[64:128]) + bh
__global__ void __launch_bounds__(256)
edge_mlp(const float* __restrict__ h,      // [N,64]
         const int* __restrict__ pairs,    // [nP,2]
         const float* __restrict__ Wh,     // [128]
         const float* __restrict__ bh,     // [1]
         float* __restrict__ out, int nP) {
  int e = blockIdx.x * blockDim.x + threadIdx.x;
  if (e >= nP) return;
  int s = pairs[2 * e + 0];
  int d = pairs[2 * e + 1];
  const float4* hs = (const float4*)(h + (size_t)s * HID);
  const float4* hd = (const float4*)(h + (size_t)d * HID);
  const float4* w0 = (const float4*)(Wh);
  const float4* w1 = (const float4*)(Wh + HID);
  float acc = bh[0];
#pragma unroll
  for (int i = 0; i < HID / 4; ++i) {
    float4 a = hs[i], w = w0[i];
    acc += a.x * w.x + a.y * w.y + a.z * w.z + a.w * w.w;
  }
#pragma unroll
  for (int i = 0; i < HID / 4; ++i) {
    float4 a = hd[i], w = w1[i];
    acc += a.x * w.x + a.y * w.y + a.z * w.z + a.w * w.w;
  }
  out[e] = acc;
}

// --------------------------------------------------------------- dispatch
extern "C" void kernel_launch(void* const* d_in, const int* in_sizes, int n_in,
                              void* d_out, int out_size, void* d_ws, size_t ws_size,
                              hipStream_t stream) {
  const float* x    = (const float*)d_in[0];
  const int*   ei   = (const int*)d_in[1];   // [2, nE]
  const int*   prs  = (const int*)d_in[2];   // [nP, 2]
  const float* Wl1  = (const float*)d_in[3];
  const float* bl1  = (const float*)d_in[4];
  const float* Wr1  = (const float*)d_in[5];
  const float* Wl2  = (const float*)d_in[6];
  const float* bl2  = (const float*)d_in[7];
  const float* Wr2  = (const float*)d_in[8];
  const float* Wh   = (const float*)d_in[9];
  const float* bh   = (const float*)d_in[10];
  float*       out  = (float*)d_out;

  const int N  = NNODES;                // 100000, divisible by 16
  const int nE = in_sizes[1] / 2;       // 3,200,000
  const int nP = in_sizes[2] / 2;       // 1,000,000

  // Workspace layout (fp32): cnt[N] | sums1[N*32] | h1[N*64] | sums2[N*64] | h2[N*64]
  float* cnt   = (float*)d_ws;
  float* sums1 = cnt + N;
  float* h1    = sums1 + (size_t)N * 32;
  float* sums2 = h1 + (size_t)N * HID;
  float* h2    = sums2 + (size_t)N * HID;

  // zero accumulators (cnt+sums1 contiguous)
  zero_f32<<<2048, 256, 0, stream>>>(cnt, (size_t)N * 33);
  zero_f32<<<2048, 256, 0, stream>>>(sums2, (size_t)N * HID);

  const int* src = ei;
  const int* dst = ei + nE;
  const int tb = 256;

  // Layer 1
  scatter_sum<32, true><<<(nE + tb - 1) / tb, tb, 0, stream>>>(x, src, dst, sums1, cnt, nE);
  sage_layer_wmma<32><<<N / 16, 128, 0, stream>>>(x, sums1, cnt, Wl1, Wr1, bl1, h1);

  // Layer 2 (degrees identical: same edge_index)
  scatter_sum<64, false><<<(nE + tb - 1) / tb, tb, 0, stream>>>(h1, src, dst, sums2, cnt, nE);
  sage_layer_wmma<64><<<N / 16, 128, 0, stream>>>(h1, sums2, cnt, Wl2, Wr2, bl2, h2);

  // Head
  edge_mlp<<<(nP + tb - 1) / tb, tb, 0, stream>>>(h2, prs, Wh, bh, out, nP);
}